// ExpertChoiceRouter_18184891532041
// MI455X (gfx1250) — compile-verified
//
#include <hip/hip_runtime.h>
#include <hip/hip_bf16.h>

// Problem constants from the reference (B=4, S=4096, D_MODEL=2048, E=64, cf=1.0)
#define D_MODEL 2048
#define TPB 256

typedef __attribute__((ext_vector_type(2))) float v2f;
typedef __attribute__((ext_vector_type(8))) float v8f;

// Monotonic float->uint key map: a > b (float)  <=>  f2k(a) > f2k(b) (uint)
__device__ __forceinline__ unsigned int f2k(float f) {
    unsigned int u = __float_as_uint(f);
    return (u & 0x80000000u) ? ~u : (u | 0x80000000u);
}
__device__ __forceinline__ float k2f(unsigned int k) {
    unsigned int u = (k & 0x80000000u) ? (k & 0x7FFFFFFFu) : ~k;
    return __uint_as_float(u);
}

// ---------------------------------------------------------------------------
// Kernel 1: affinity[e][n] = sum_d w_sel[e][d] * tokens[n][d]
// One wave computes a 16(expert) x 16(token) tile via V_WMMA_F32_16X16X4_F32.
// Block = 256 threads = 8 waves: 2 token tiles x 4 expert tiles (covers E=64).
// ---------------------------------------------------------------------------
__global__ void __launch_bounds__(TPB)
affinity_gemm_wmma(const float* __restrict__ tokens,
                   const float* __restrict__ wsel,
                   float* __restrict__ affinity, int N) {
    const int lane = threadIdx.x & 31;
    const int wave = threadIdx.x >> 5;
    const int half = lane >> 4;   // 0: lanes 0-15, 1: lanes 16-31
    const int row  = lane & 15;

    const int n0 = blockIdx.x * 32 + (wave & 1) * 16;   // token tile base
    const int e0 = (wave >> 1) * 16;                    // expert tile base

    // A (16x4 f32): lanes 0-15 carry K=0,1 for M=row; lanes 16-31 carry K=2,3
    const float* __restrict__ arow = wsel   + (size_t)(e0 + row) * D_MODEL + 2 * half;
    // B (4x16 f32): symmetric layout, N=row across lanes, K split by lane half
    const float* __restrict__ brow = tokens + (size_t)(n0 + row) * D_MODEL + 2 * half;

    v8f acc = {};
#pragma unroll 8
    for (int d = 0; d < D_MODEL; d += 4) {
        v2f a = *(const v2f*)(arow + d);
        v2f b = *(const v2f*)(brow + d);
        // (neg_a, A, neg_b, B, c_mod, C, reuse_a, reuse_b)
        acc = __builtin_amdgcn_wmma_f32_16x16x4_f32(
            false, a, false, b, (short)0, acc, false, false);
    }

    // C/D layout: VGPR r holds M = r + 8*half, N = row
#pragma unroll
    for (int r = 0; r < 8; ++r) {
        const int e = e0 + r + 8 * half;
        affinity[(size_t)e * N + n0 + row] = acc[r];
    }
}

// ---------------------------------------------------------------------------
// Kernel 2: per-expert top-k (k=cap) + softmax, scattered in place.
// One block of 256 threads per expert; thread t owns the contiguous chunk
// [t*chunk, (t+1)*chunk) so tie ranks follow token-index order (matches
// jax.lax.top_k tie-breaking: lowest index first).
// ---------------------------------------------------------------------------
__global__ void __launch_bounds__(TPB)
expert_topk_softmax(float* __restrict__ aff, int N, int k) {
    __shared__ unsigned int hist[256];
    __shared__ float        redf[TPB];
    __shared__ unsigned int scanbuf[TPB];
    __shared__ unsigned int s_bin, s_rem;
    __shared__ float        s_max, s_Z;

    const int t = threadIdx.x;
    float* __restrict__ rowp = aff + (size_t)blockIdx.x * N;
    const int chunk = N / TPB;      // 64 for N=16384
    const int base  = t * chunk;

    // ---- row max (= top-1 score, softmax shift) ----
    float m = -3.402823466e38f;
    for (int i = 0; i < chunk; ++i) m = fmaxf(m, rowp[base + i]);
    redf[t] = m;
    __syncthreads();
    for (int s = TPB / 2; s > 0; s >>= 1) {
        if (t < s) redf[t] = fmaxf(redf[t], redf[t + s]);
        __syncthreads();
    }
    if (t == 0) s_max = redf[0];
    __syncthreads();
    const float M = s_max;

    // ---- radix-select the k-th largest key (exact 32-bit threshold) ----
    unsigned int prefix = 0, maskHi = 0, remaining = (unsigned int)k;
    for (int shift = 24; shift >= 0; shift -= 8) {
        hist[t] = 0;                          // TPB == 256 bins
        __syncthreads();
        for (int i = 0; i < chunk; ++i) {
            unsigned int key = f2k(rowp[base + i]);
            if ((key & maskHi) == prefix)
                atomicAdd(&hist[(key >> shift) & 255u], 1u);
        }
        __syncthreads();
        if (t == 0) {
            unsigned int cum = 0;
            for (int b = 255; b >= 0; --b) {
                unsigned int c = hist[b];
                if (cum + c >= remaining) { s_bin = (unsigned int)b; s_rem = remaining - cum; break; }
                cum += c;
            }
        }
        __syncthreads();
        prefix |= (s_bin << shift);
        maskHi |= (0xFFu << shift);
        remaining = s_rem;
        __syncthreads();
    }
    const unsigned int T    = prefix;     // key of the k-th largest value
    const unsigned int need = remaining;  // #ties at T that make the cut
    const float vT = k2f(T);

    // ---- softmax normalizer over the selected set ----
    float z = 0.0f;
    for (int i = 0; i < chunk; ++i) {
        float v = rowp[base + i];
        if (f2k(v) > T) z += __expf(v - M);
    }
    redf[t] = z;
    __syncthreads();
    for (int s = TPB / 2; s > 0; s >>= 1) {
        if (t < s) redf[t] += redf[t + s];
        __syncthreads();
    }
    if (t == 0) s_Z = redf[0] + (float)need * __expf(vT - M);
    __syncthreads();
    const float Z = s_Z;

    // ---- tie ranks by token index (exclusive scan of per-chunk eq counts) ----
    unsigned int eqc = 0;
    for (int i = 0; i < chunk; ++i)
        if (f2k(rowp[base + i]) == T) eqc++;
    scanbuf[t] = eqc;
    __syncthreads();
    if (t == 0) {
        unsigned int acc = 0;
        for (int i = 0; i < TPB; ++i) { unsigned int c = scanbuf[i]; scanbuf[i] = acc; acc += c; }
    }
    __syncthreads();
    unsigned int rank = scanbuf[t];

    // ---- scatter softmax weights in place (0 = not selected) ----
    const float wT = __expf(vT - M) / Z;
    for (int i = 0; i < chunk; ++i) {
        float v = rowp[base + i];
        unsigned int key = f2k(v);
        float w = 0.0f;
        if (key > T) {
            w = __expf(v - M) / Z;
        } else if (key == T) {
            if (rank < need) w = wT;
            rank++;
        }
        rowp[base + i] = w;
    }
}

// ---------------------------------------------------------------------------
// Kernel 3: counts = clip(#experts per token, 1); transpose [E][N] -> [N][E];
// emit expert_weights, expert_assignments, capacity scalar.
// ---------------------------------------------------------------------------
__global__ void __launch_bounds__(TPB)
finalize_outputs(const float* __restrict__ wT, float* __restrict__ out,
                 int N, int E, int out_size, float capf) {
    const int n = blockIdx.x * blockDim.x + threadIdx.x;
    if (n >= N) return;

    float cnt = 0.0f;
    for (int e = 0; e < E; ++e)
        cnt += (wT[(size_t)e * N + n] != 0.0f) ? 1.0f : 0.0f;
    const float inv = 1.0f / fmaxf(cnt, 1.0f);

    float* __restrict__ ow = out + (size_t)n * E;                   // expert_weights [N][E]
    float* __restrict__ oa = out + (size_t)N * E + (size_t)n * E;   // expert_assignments [N][E]
    for (int e = 0; e < E; ++e) {
        const float w = wT[(size_t)e * N + n];
        ow[e] = w * inv;
        oa[e] = (w != 0.0f) ? 1.0f : 0.0f;
    }

    if (n == 0 && out_size > 2 * N * E)
        out[(size_t)2 * N * E] = capf;   // capacity (python int -> trailing scalar)
}

extern "C" void kernel_launch(void* const* d_in, const int* in_sizes, int n_in,
                              void* d_out, int out_size, void* d_ws, size_t ws_size,
                              hipStream_t stream) {
    const float* tokens = (const float*)d_in[0];   // [B,S,D] fp32, viewed as [N,D]
    const float* wsel   = (const float*)d_in[1];   // [E,D]  fp32

    const int N   = in_sizes[0] / D_MODEL;   // 16384
    const int E   = in_sizes[1] / D_MODEL;   // 64
    const int cap = N / E;                   // 256 (CAPACITY_FACTOR = 1.0)

    float* aff = (float*)d_ws;               // E*N*4 = 4 MB scratch, fully overwritten

    // 1) affinity GEMM via f32 WMMA: 512 blocks x 8 waves, tile 16x16
    affinity_gemm_wmma<<<N / 32, TPB, 0, stream>>>(tokens, wsel, aff, N);

    // 2) per-expert top-cap + softmax (in place)
    expert_topk_softmax<<<E, TPB, 0, stream>>>(aff, N, cap);

    // 3) counts, transpose, outputs
    finalize_outputs<<<(N + TPB - 1) / TPB, TPB, 0, stream>>>(
        aff, (float*)d_out, N, E, out_size, (float)cap);
}